// MultiBoxLoss_50002009260496
// MI455X (gfx1250) — compile-verified
//
#include <hip/hip_runtime.h>
#include <hip/hip_bf16.h>

#define NB     8732
#define NC     21
#define NBATCH 128

typedef float v2f __attribute__((ext_vector_type(2)));
typedef float v8f __attribute__((ext_vector_type(8)));

// ---------------------------------------------------------------------------
// Stage 1: one workgroup per batch row.
// Phase A (WMMA): per-box softmax denominator s and per-box max m into LDS.
//   Wave handles 16 boxes; lane L (<16) and lane L+16 hold class pairs
//   (k0,k0+1) with k0 = 4c + 2*hi, matching the 16x4 f32 A-matrix layout.
//   B = ones => D row-sums; 6 WMMAs cover 24 (21 padded) classes.
// Phase B: per-box ce = log(s)+m - gold; accumulate per-row num_pos,
//   ce over positives, smooth-L1 loc loss over positives.
// ---------------------------------------------------------------------------
__global__ __launch_bounds__(256) void ssd_stage1(
    const float* __restrict__ locP, const float* __restrict__ locT,
    const float* __restrict__ conf, const int* __restrict__ lbl,
    float* __restrict__ ceOut, float* __restrict__ rowLoc,
    float* __restrict__ rowCe, int* __restrict__ rowNp)
{
    __shared__ float lsS[NB];
    __shared__ float lsM[NB];
    __shared__ float rF0[256];
    __shared__ float rF1[256];
    __shared__ int   rI0[256];

    const int row  = blockIdx.x;
    const int tid  = threadIdx.x;
    const int wave = tid >> 5;
    const int lane = tid & 31;
    const int mrow = lane & 15;
    const int hi   = lane >> 4;

    const float* cbase = conf + (size_t)row * NB * NC;

    const int ngroups = (NB + 15) / 16;  // 546
    for (int g = wave; g < ngroups; g += 8) {
        int  bx     = g * 16 + mrow;
        bool bvalid = bx < NB;
        int  bc     = bvalid ? bx : (NB - 1);
        const float* p = cbase + (size_t)bc * NC;

        float va[6], vb[6];
#pragma unroll
        for (int c = 0; c < 6; ++c) {
            int k0 = 4 * c + 2 * hi;
            int ka = (k0     < NC) ? k0     : NC - 1;
            int kb = (k0 + 1 < NC) ? k0 + 1 : NC - 1;
            float a = p[ka];
            float b = p[kb];
            va[c] = (bvalid && k0     < NC) ? a : -1e30f;
            vb[c] = (bvalid && k0 + 1 < NC) ? b : -1e30f;
        }
        // per-box max: reduce 12 local values, then swap halves (wave32)
        float m = va[0];
#pragma unroll
        for (int c = 0; c < 6; ++c) { m = fmaxf(m, va[c]); m = fmaxf(m, vb[c]); }
        m = fmaxf(m, __shfl_xor(m, 16, 32));

        // 6 x V_WMMA_F32_16X16X4_F32 with B = ones -> per-box exp-sums
        v8f acc = {};
#pragma unroll
        for (int c = 0; c < 6; ++c) {
            v2f A;  A.x = __expf(va[c] - m); A.y = __expf(vb[c] - m);
            v2f Bv; Bv.x = 1.0f; Bv.y = 1.0f;
            acc = __builtin_amdgcn_wmma_f32_16x16x4_f32(
                false, A, false, Bv, (short)0, acc, false, false);
        }

        if (hi == 0 && bvalid) lsM[bx] = m;
        // D layout: VGPR r holds box r (lanes 0-15) / box 8+r (lanes 16-31)
#pragma unroll
        for (int r = 0; r < 8; ++r) {
            int bxo = g * 16 + r + 8 * hi;
            if (mrow == r && bxo < NB) lsS[bxo] = acc[r];
        }
    }
    __syncthreads();

    // Phase B: per-box epilogue + per-row partials
    float cePos = 0.f, locAcc = 0.f;
    int   np = 0;
    const size_t rb = (size_t)row * NB;
    for (int i = tid; i < NB; i += 256) {
        float lse = __logf(lsS[i]) + lsM[i];
        int   L   = lbl[rb + i];
        int   Lc  = (L < 0) ? 0 : ((L >= NC) ? NC - 1 : L);
        float gold = cbase[(size_t)i * NC + Lc];
        float ce   = lse - gold;
        ceOut[rb + i] = ce;
        if (L > 0) {
            np++;
            cePos += ce;
            const float* lp = locP + (rb + i) * 4;
            const float* lt = locT + (rb + i) * 4;
            float s = 0.f;
#pragma unroll
            for (int k = 0; k < 4; ++k) {
                float d  = lp[k] - lt[k];
                float ad = fabsf(d);
                s += (ad < 1.f) ? 0.5f * d * d : (ad - 0.5f);
            }
            locAcc += s;
        }
    }
    rF0[tid] = cePos; rF1[tid] = locAcc; rI0[tid] = np;
    __syncthreads();
    for (int off = 128; off > 0; off >>= 1) {
        if (tid < off) {
            rF0[tid] += rF0[tid + off];
            rF1[tid] += rF1[tid + off];
            rI0[tid] += rI0[tid + off];
        }
        __syncthreads();
    }
    if (tid == 0) {
        rowCe[row]  = rF0[0];
        rowLoc[row] = rF1[0];
        rowNp[row]  = rI0[0];
    }
}

// ---------------------------------------------------------------------------
// Stage 2: one workgroup per row. Top-K sum of negative ce via bitwise
// threshold search on float bits (monotone for ce >= 0). Ties at the
// threshold contribute (K - cnt_gt) * T exactly, so no stable sort needed.
// ---------------------------------------------------------------------------
__global__ __launch_bounds__(256) void ssd_stage2(
    const float* __restrict__ ceIn, const int* __restrict__ lbl,
    const int* __restrict__ rowNp, float* __restrict__ rowNeg)
{
    __shared__ unsigned keys[NB];
    __shared__ int   sCnt;
    __shared__ float rF[256];
    __shared__ int   rI[256];

    const int row = blockIdx.x;
    const int tid = threadIdx.x;
    const size_t rb = (size_t)row * NB;

    for (int i = tid; i < NB; i += 256) {
        int   L  = lbl[rb + i];
        float ce = ceIn[rb + i];
        keys[i] = (L > 0) ? 0u : __float_as_uint(fmaxf(ce, 0.f));
    }
    __syncthreads();

    int np = rowNp[row];
    int K  = 3 * np;
    if (K > NB - 1) K = NB - 1;

    float result = 0.f;
    if (K > 0) {
        unsigned T = 0u;
        for (int bit = 31; bit >= 0; --bit) {
            unsigned trial = T | (1u << bit);
            if (tid == 0) sCnt = 0;
            __syncthreads();
            int c = 0;
            for (int i = tid; i < NB; i += 256) c += (keys[i] >= trial) ? 1 : 0;
            atomicAdd(&sCnt, c);
            __syncthreads();
            if (sCnt >= K) T = trial;   // uniform across threads
            __syncthreads();
        }
        float s = 0.f; int cgt = 0;
        for (int i = tid; i < NB; i += 256) {
            unsigned k = keys[i];
            if (k > T) { s += __uint_as_float(k); cgt++; }
        }
        rF[tid] = s; rI[tid] = cgt;
        __syncthreads();
        for (int off = 128; off > 0; off >>= 1) {
            if (tid < off) { rF[tid] += rF[tid + off]; rI[tid] += rI[tid + off]; }
            __syncthreads();
        }
        result = rF[0] + (float)(K - rI[0]) * __uint_as_float(T);
    }
    if (tid == 0) rowNeg[row] = result;
}

// ---------------------------------------------------------------------------
// Stage 3: single block, deterministic combine of 128 per-row partials.
// ---------------------------------------------------------------------------
__global__ __launch_bounds__(128) void ssd_stage3(
    const float* __restrict__ rowLoc, const float* __restrict__ rowCe,
    const float* __restrict__ rowNeg, const int* __restrict__ rowNp,
    float* __restrict__ out)
{
    __shared__ float rF[128];
    __shared__ int   rI[128];
    int tid = threadIdx.x;
    rF[tid] = rowLoc[tid] + rowCe[tid] + rowNeg[tid];
    rI[tid] = rowNp[tid];
    __syncthreads();
    for (int off = 64; off > 0; off >>= 1) {
        if (tid < off) { rF[tid] += rF[tid + off]; rI[tid] += rI[tid + off]; }
        __syncthreads();
    }
    if (tid == 0) {
        float denom = (float)rI[0];
        if (denom < 1.f) denom = 1.f;
        out[0] = rF[0] / denom;
    }
}

extern "C" void kernel_launch(void* const* d_in, const int* in_sizes, int n_in,
                              void* d_out, int out_size, void* d_ws, size_t ws_size,
                              hipStream_t stream)
{
    const float* locP = (const float*)d_in[0];
    const float* locT = (const float*)d_in[1];
    const float* conf = (const float*)d_in[2];
    const int*   lbl  = (const int*)d_in[3];

    float* ws     = (float*)d_ws;
    float* ceOut  = ws;                                // 128*8732 floats
    float* rowLoc = ws + (size_t)NBATCH * NB;
    float* rowCe  = rowLoc + NBATCH;
    float* rowNeg = rowCe + NBATCH;
    int*   rowNp  = (int*)(rowNeg + NBATCH);

    ssd_stage1<<<NBATCH, 256, 0, stream>>>(locP, locT, conf, lbl,
                                           ceOut, rowLoc, rowCe, rowNp);
    ssd_stage2<<<NBATCH, 256, 0, stream>>>(ceOut, lbl, rowNp, rowNeg);
    ssd_stage3<<<1, 128, 0, stream>>>(rowLoc, rowCe, rowNeg, rowNp,
                                      (float*)d_out);
}